// LocalNodeVariant_t_6777458393750
// MI455X (gfx1250) — compile-verified
//
#include <hip/hip_runtime.h>
#include <hip/hip_bf16.h>

// ---------------------------------------------------------------------------
// LocalNodeVariant GNN forward for MI455X (gfx1250, wave32, WMMA)
//
// Dominant cost: 4 graph-shift GEMMs against S (4096x4096 fp32, L2-resident,
// ~172 GFLOP fp32 total). They run on the native fp32 matrix pipe
// (v_wmma_f32_16x16x4_f32) with double-buffered async global->LDS staging
// (global_load_async_to_lds_b128 + s_wait_asynccnt).
// ---------------------------------------------------------------------------

typedef __attribute__((ext_vector_type(2))) float v2f;
typedef __attribute__((ext_vector_type(4))) float f4;
typedef __attribute__((ext_vector_type(8))) float v8f;

#define NN 4096          // number of graph nodes (= GEMM M and K)
#define KC 32            // K-chunk (n) per LDS stage
#define LDSP 36          // padded LDS row stride (floats): 36r mod 64 distinct
#define NCHUNK (NN / KC) // 128

// LDS generic pointers: low 32 bits are the LDS segment byte offset
// (ISA 10.2).
__device__ __forceinline__ uint32_t lds_off(const void* p) {
    return (uint32_t)(uintptr_t)p;
}

// Async DMA: global (saddr + voffset) -> LDS byte offset in ldsoff, 16 bytes per lane.
// Tracked with ASYNCcnt; no VGPR round-trip, no per-load loadcnt waits.
__device__ __forceinline__ void async_ld128(uint32_t ldsoff, uint32_t goff,
                                            const float* base) {
    asm volatile("global_load_async_to_lds_b128 %0, %1, %2"
                 :: "v"(ldsoff), "v"(goff), "s"(base)
                 : "memory");
}

__device__ __forceinline__ void wait_async0() {
    asm volatile("s_wait_asynccnt 0x0" ::: "memory");
}

// ---------------------------------------------------------------------------
// Shift GEMM:  C[r, m] = sum_n X[r, n] * S[m, n]      (C = X * S^T)
//   X: [R, 4096] row-major, S: [4096, 4096] row-major, C: [R, 4096] row-major
//   grid.x = 4096/128 (m tiles), grid.y = R/64 (r tiles), block = 256 (8 waves)
//   Each wave: 16 m-columns, 4 stacked 16-row fragments (64r x 16m per wave).
// ---------------------------------------------------------------------------
__global__ __launch_bounds__(256) void shift_gemm(const float* __restrict__ X,
                                                  const float* __restrict__ S,
                                                  float* __restrict__ C)
{
    __shared__ float ldsX[2][64 * LDSP];    // 64 r-rows  x 32 n-cols, x2 bufs
    __shared__ float ldsS[2][128 * LDSP];   // 128 m-rows x 32 n-cols, x2 bufs

    const int tid  = threadIdx.x;
    const int lane = tid & 31;
    const int wave = tid >> 5;           // 0..7 -> m sub-tile
    const int m0   = blockIdx.x * 128;
    const int r0   = blockIdx.y * 64;

    const int lrow  = tid >> 3;          // 0..31 (loader row group)
    const int lcol4 = (tid & 7) * 4;     // 0,4,...,28 (loader col, float4)

    const int half = lane >> 4;          // 0: K pair {0,1}, 1: K pair {2,3}
    const int rsel = lane & 15;          // fragment row / column select

    // per-thread global byte offsets for chunk 0
    const uint32_t xoff0 = (uint32_t)(((r0 + lrow)      * NN + lcol4) * 4);
    const uint32_t xoff1 = (uint32_t)(((r0 + lrow + 32) * NN + lcol4) * 4);
    const uint32_t soff0 = (uint32_t)(((m0 + lrow)      * NN + lcol4) * 4);
    const uint32_t soff1 = (uint32_t)(((m0 + lrow + 32) * NN + lcol4) * 4);
    const uint32_t soff2 = (uint32_t)(((m0 + lrow + 64) * NN + lcol4) * 4);
    const uint32_t soff3 = (uint32_t)(((m0 + lrow + 96) * NN + lcol4) * 4);

    // per-thread LDS byte offsets (buffer 0); buffer/row strides in bytes
    const uint32_t xlds = lds_off(&ldsX[0][lrow * LDSP + lcol4]);
    const uint32_t slds = lds_off(&ldsS[0][lrow * LDSP + lcol4]);
    const uint32_t XBUF = 64  * LDSP * 4;   // ldsX buffer stride
    const uint32_t SBUF = 128 * LDSP * 4;   // ldsS buffer stride
    const uint32_t R32  = 32  * LDSP * 4;   // 32-row stride

    v8f c0 = {}, c1 = {}, c2 = {}, c3 = {};

    // prologue: stage chunk 0 into buffer 0
    async_ld128(xlds,           xoff0, X);
    async_ld128(xlds + R32,     xoff1, X);
    async_ld128(slds,           soff0, S);
    async_ld128(slds + R32,     soff1, S);
    async_ld128(slds + 2 * R32, soff2, S);
    async_ld128(slds + 3 * R32, soff3, S);

    for (int c = 0; c < NCHUNK; ++c) {
        wait_async0();          // this wave's DMA into buf[c&1] landed
        __syncthreads();        // all waves' DMA landed; prev compute done

        // stage chunk c+1 into the other buffer while we compute chunk c
        if (c + 1 < NCHUNK) {
            const uint32_t nb = (uint32_t)((c + 1) * KC * 4);
            const uint32_t xb = xlds + (uint32_t)((c + 1) & 1) * XBUF;
            const uint32_t sb = slds + (uint32_t)((c + 1) & 1) * SBUF;
            async_ld128(xb,           xoff0 + nb, X);
            async_ld128(xb + R32,     xoff1 + nb, X);
            async_ld128(sb,           soff0 + nb, S);
            async_ld128(sb + R32,     soff1 + nb, S);
            async_ld128(sb + 2 * R32, soff2 + nb, S);
            async_ld128(sb + 3 * R32, soff3 + nb, S);
        }

        const float* bX = &ldsX[c & 1][0];
        const float* bS = &ldsS[c & 1][0];

        // 8 K-steps of 4 over this chunk, 4 stacked row fragments per wave
        #pragma unroll
        for (int kk = 0; kk < 8; ++kk) {
            const int kb = kk * 4 + half * 2;

            // B fragment: S^T tile, element (k, mcol) = S[m0+mcol, n0+k]
            v2f b;
            b.x = bS[(wave * 16 + rsel) * LDSP + kb];
            b.y = bS[(wave * 16 + rsel) * LDSP + kb + 1];

            v2f a0, a1, a2, a3;
            a0.x = bX[(0 * 16 + rsel) * LDSP + kb];
            a0.y = bX[(0 * 16 + rsel) * LDSP + kb + 1];
            a1.x = bX[(1 * 16 + rsel) * LDSP + kb];
            a1.y = bX[(1 * 16 + rsel) * LDSP + kb + 1];
            a2.x = bX[(2 * 16 + rsel) * LDSP + kb];
            a2.y = bX[(2 * 16 + rsel) * LDSP + kb + 1];
            a3.x = bX[(3 * 16 + rsel) * LDSP + kb];
            a3.y = bX[(3 * 16 + rsel) * LDSP + kb + 1];

            c0 = __builtin_amdgcn_wmma_f32_16x16x4_f32(false, a0, false, b,
                                                       (short)0, c0, false, false);
            c1 = __builtin_amdgcn_wmma_f32_16x16x4_f32(false, a1, false, b,
                                                       (short)0, c1, false, false);
            c2 = __builtin_amdgcn_wmma_f32_16x16x4_f32(false, a2, false, b,
                                                       (short)0, c2, false, false);
            c3 = __builtin_amdgcn_wmma_f32_16x16x4_f32(false, a3, false, b,
                                                       (short)0, c3, false, false);
        }
        __syncthreads();        // done reading buf[c&1] before it is re-staged
    }

    // --- write back: c[i] holds (M = i + 8*half, N = rsel) -----------------
    const int col = m0 + wave * 16 + rsel;
    #pragma unroll
    for (int i = 0; i < 8; ++i) {
        const int row = i + 8 * half;
        C[(size_t)(r0 + 0 * 16 + row) * NN + col] = c0[i];
        C[(size_t)(r0 + 1 * 16 + row) * NN + col] = c1[i];
        C[(size_t)(r0 + 2 * 16 + row) * NN + col] = c2[i];
        C[(size_t)(r0 + 3 * 16 + row) * NN + col] = c3[i];
    }
}

// ---------------------------------------------------------------------------
// Node-variant tap combine:
//   y[bt, f, n] = tanh( b[f] + sum_{k,g} z_k[bt, g, n] * h[f, k, n, g] )
// One thread per (bt, n). h is contiguous in g -> dense vector loads.
// ---------------------------------------------------------------------------
template <int G>
__global__ __launch_bounds__(256) void tap_combine(const float* __restrict__ z0,
                                                   const float* __restrict__ z1,
                                                   const float* __restrict__ z2,
                                                   const float* __restrict__ h,
                                                   const float* __restrict__ bias,
                                                   float* __restrict__ y)
{
    const int idx = blockIdx.x * blockDim.x + threadIdx.x;
    const int bt  = idx >> 12;           // / 4096
    const int n   = idx & (NN - 1);

    float zv[3][G];
    #pragma unroll
    for (int g = 0; g < G; ++g) {
        const size_t zi = ((size_t)bt * G + g) * NN + n;
        zv[0][g] = z0[zi];
        zv[1][g] = z1[zi];
        zv[2][g] = z2[zi];
    }

    #pragma unroll
    for (int f = 0; f < 16; ++f) {
        float acc = bias[f];
        #pragma unroll
        for (int k = 0; k < 3; ++k) {
            const float* hp = h + (((size_t)(f * 3 + k)) * NN + n) * G;
            #pragma unroll
            for (int g = 0; g < G; ++g)
                acc += hp[g] * zv[k][g];
        }
        y[((size_t)bt * 16 + f) * NN + n] = tanhf(acc);
    }
}

// ---------------------------------------------------------------------------
// Per-node readout MLP: out = W2 * tanh(W1 * y + bw1) + bw2
// One thread per (bt, n). Weights staged in LDS.
// ---------------------------------------------------------------------------
__global__ __launch_bounds__(256) void readout(const float* __restrict__ y2,
                                               const float* __restrict__ W1,
                                               const float* __restrict__ bw1,
                                               const float* __restrict__ W2,
                                               const float* __restrict__ bw2,
                                               float* __restrict__ out)
{
    __shared__ float sW1[256];
    __shared__ float sW2[64];
    __shared__ float sb1[16];
    __shared__ float sb2[4];

    const int tid = threadIdx.x;
    sW1[tid] = W1[tid];
    if (tid < 64) sW2[tid] = W2[tid];
    if (tid < 16) sb1[tid] = bw1[tid];
    if (tid < 4)  sb2[tid] = bw2[tid];
    __syncthreads();

    const int idx = blockIdx.x * blockDim.x + tid;
    const int bt  = idx >> 12;
    const int n   = idx & (NN - 1);

    float v[16];
    #pragma unroll
    for (int f = 0; f < 16; ++f)
        v[f] = y2[((size_t)bt * 16 + f) * NN + n];

    float r1[16];
    #pragma unroll
    for (int j = 0; j < 16; ++j) {
        float acc = sb1[j];
        #pragma unroll
        for (int i = 0; i < 16; ++i)
            acc += sW1[j * 16 + i] * v[i];
        r1[j] = tanhf(acc);
    }

    #pragma unroll
    for (int o = 0; o < 4; ++o) {
        float acc = sb2[o];
        #pragma unroll
        for (int j = 0; j < 16; ++j)
            acc += sW2[o * 16 + j] * r1[j];
        out[((size_t)bt * 4 + o) * NN + n] = acc;
    }
}

// ---------------------------------------------------------------------------
extern "C" void kernel_launch(void* const* d_in, const int* in_sizes, int n_in,
                              void* d_out, int out_size, void* d_ws, size_t ws_size,
                              hipStream_t stream)
{
    (void)in_sizes; (void)n_in; (void)out_size; (void)ws_size;

    const float* x   = (const float*)d_in[0];  // [4,32,4,4096]   -> [512, 4096]
    const float* S   = (const float*)d_in[1];  // [4096, 4096]
    const float* h1  = (const float*)d_in[2];  // [16,3,4096,4]
    const float* b1  = (const float*)d_in[3];  // [16,1]
    const float* h2  = (const float*)d_in[4];  // [16,3,4096,16]
    const float* b2  = (const float*)d_in[5];  // [16,1]
    const float* W1  = (const float*)d_in[6];  // [16,16]
    const float* bw1 = (const float*)d_in[7];  // [16]
    const float* W2  = (const float*)d_in[8];  // [4,16]
    const float* bw2 = (const float*)d_in[9];  // [4]
    float* out = (float*)d_out;                // [4,32,4,4096]

    // workspace carve-up (fp32 elements)
    float* ws  = (float*)d_ws;
    float* z1a = ws;                             //  512*4096  ( 8 MB)
    float* z2a = z1a + (size_t)512 * NN;         //  512*4096  ( 8 MB)
    float* y1  = z2a + (size_t)512 * NN;         // 2048*4096  (32 MB)
    float* z1b = y1  + (size_t)2048 * NN;        // 2048*4096  (32 MB)
    float* z2b = z1b + (size_t)2048 * NN;        // 2048*4096  (32 MB)
    float* y2  = z2b + (size_t)2048 * NN;        // 2048*4096  (32 MB)

    const dim3 blk(256);

    // Layer 1: shifts (R = B*T*F0 = 512 rows), taps (G = 4)
    shift_gemm<<<dim3(NN / 128, 512 / 64), blk, 0, stream>>>(x,   S, z1a);
    shift_gemm<<<dim3(NN / 128, 512 / 64), blk, 0, stream>>>(z1a, S, z2a);
    tap_combine<4><<<(128 * NN) / 256, blk, 0, stream>>>(x, z1a, z2a, h1, b1, y1);

    // Layer 2: shifts (R = B*T*F1 = 2048 rows), taps (G = 16)
    shift_gemm<<<dim3(NN / 128, 2048 / 64), blk, 0, stream>>>(y1,  S, z1b);
    shift_gemm<<<dim3(NN / 128, 2048 / 64), blk, 0, stream>>>(z1b, S, z2b);
    tap_combine<16><<<(128 * NN) / 256, blk, 0, stream>>>(y1, z1b, z2b, h2, b2, y2);

    // Per-node readout MLP
    readout<<<(128 * NN) / 256, blk, 0, stream>>>(y2, W1, bw1, W2, bw2, out);
}